// LargeMarginSoftmaxV2_45526653337749
// MI455X (gfx1250) — compile-verified
//
#include <hip/hip_runtime.h>

// ---------------------------------------------------------------------------
// LargeMarginSoftmaxV2 loss, MI455X (gfx1250, wave32).
// logits: (8, 19, 512, 512) f32, labels: (8, 512, 512) i32 -> scalar f32.
//
// Memory-bound streaming kernel (~167MB @ 23.3TB/s ~ 7us floor); per-pixel
// math minimized via log-sum cancellation (see analysis). Wave-level
// reduction of the (loss, count) pair is done with one v_wmma_f32_16x16x4_f32.
// ---------------------------------------------------------------------------

typedef float v2f __attribute__((ext_vector_type(2)));
typedef float v8f __attribute__((ext_vector_type(8)));

#define NCLS 19
#define HW_SHIFT 18
#define HW (1 << HW_SHIFT)          // 512*512
#define PIX_TOTAL (8 * HW)          // 2,097,152 pixels
#define TPB 256
#define PIX_PER_THREAD 4
#define NBLOCKS (PIX_TOTAL / (TPB * PIX_PER_THREAD))   // 2048

__device__ __forceinline__ float comp4(const float4& v, int j) {
  return (j == 0) ? v.x : (j == 1) ? v.y : (j == 2) ? v.z : v.w;
}

// Reduce (loss, cnt) across the 32 lanes of a wave with one f32 WMMA.
// A[m,k] holds the per-lane pair (a0=loss, a1=cnt); B is a 0/1 column
// selector so D columns 0-7 carry loss row-sums and columns 8-15 carry
// count row-sums. One shfl_xor(16) then completes both totals at once.
__device__ __forceinline__ void wave_reduce_pair_wmma(float loss, float cnt,
                                                      float* loss_tot,
                                                      float* cnt_tot) {
  const unsigned lane = threadIdx.x & 31u;
  v2f a;
  a[0] = loss;                      // A[m,0] (lanes 0-15) / A[m,2] (lanes 16-31)
  a[1] = cnt;                       // A[m,1]              / A[m,3]
  const float sel = ((lane & 15u) < 8u) ? 1.0f : 0.0f;
  v2f b;
  b[0] = sel;                       // B rows 0,2: pick loss into columns 0-7
  b[1] = 1.0f - sel;                // B rows 1,3: pick cnt into columns 8-15
  v8f c = {};
  v8f d = __builtin_amdgcn_wmma_f32_16x16x4_f32(
      /*neg_a=*/false, a, /*neg_b=*/false, b,
      /*c_mod=*/(short)0, c, /*reuse_a=*/false, /*reuse_b=*/false);
  float s = d[0] + d[1] + d[2] + d[3] + d[4] + d[5] + d[6] + d[7];
  s += __shfl_xor(s, 16, 32);       // fold the two M-halves together
  *loss_tot = __shfl(s, 0, 32);     // any lane 0-7 holds the loss total
  *cnt_tot = __shfl(s, 8, 32);      // any lane 8-15 holds the count total
}

__global__ __launch_bounds__(TPB) void lms_pixel_kernel(
    const float* __restrict__ logits, const int* __restrict__ labels,
    float2* __restrict__ partials) {
  const unsigned tid = blockIdx.x * TPB + threadIdx.x;
  const unsigned pix0 = tid * PIX_PER_THREAD;      // 16B-aligned pixel group
  const unsigned n = pix0 >> HW_SHIFT;
  const unsigned hw = pix0 & (HW - 1);
  const float* base = logits + ((size_t)n * NCLS << HW_SHIFT) + hw;

  float4 x[NCLS];
#pragma unroll
  for (int c = 0; c < NCLS; ++c)
    x[c] = *reinterpret_cast<const float4*>(base + ((size_t)c << HW_SHIFT));
  const int4 lb4 = *reinterpret_cast<const int4*>(labels + pix0);

  const float COEFF = 1.0f / (float)(NCLS - 1);
  const float LAMH = 0.15f;         // LAM / 2
  const float NEGBIG = -3.0e38f;

  float loss_acc = 0.0f, cnt_acc = 0.0f;
#pragma unroll
  for (int j = 0; j < PIX_PER_THREAD; ++j) {
    const int lbl = (j == 0) ? lb4.x : (j == 1) ? lb4.y : (j == 2) ? lb4.z
                                                                   : lb4.w;
    const bool valid = (lbl != 255);
    const int l = valid ? lbl : 0;

    // Pass 1 over registers: masked max M2 (label excluded) and x_label.
    float M2 = NEGBIG, xl = 0.0f;
#pragma unroll
    for (int c = 0; c < NCLS; ++c) {
      const float xc = comp4(x[c], j);
      const bool isl = (c == l);
      xl = isl ? xc : xl;
      M2 = fmaxf(M2, isl ? NEGBIG : xc);
    }

    // Pass 2: S2 = sum_{c!=l} e^(d), A = sum e*d, B = sum d  (d = x_c - M2).
    float S2 = 0.0f, A = 0.0f, B = 0.0f;
#pragma unroll
    for (int c = 0; c < NCLS; ++c) {
      const float xc = comp4(x[c], j);
      const bool isl = (c == l);
      float dd = xc - M2;
      float e = __expf(dd);
      e = isl ? 0.0f : e;
      dd = isl ? 0.0f : dd;
      S2 += e;
      A = fmaf(e, dd, A);
      B += dd;
    }

    // margin = lam * sum_{c!=l} (q - coeff) * log q ; the log(S2) terms
    // cancel exactly because coeff*(C-1) == 1:
    //   margin = lam * (A/S2 - coeff*B)
    const float margin = LAMH * (A / S2 - COEFF * B);

    // Full-softmax CE: M = max(M2, x_l); S = e^(x_l-M) + e^(M2-M)*S2.
    const float M = fmaxf(M2, xl);
    const float S = __expf(xl - M) + __expf(M2 - M) * S2;
    const float ce = __logf(S) - (xl - M);

    loss_acc += valid ? (ce + margin) : 0.0f;
    cnt_acc += valid ? 1.0f : 0.0f;
  }

  // Wave reduction via WMMA (EXEC all-ones: grid covers pixels exactly).
  float lt, ct;
  wave_reduce_pair_wmma(loss_acc, cnt_acc, &lt, &ct);

  __shared__ float2 wsum[TPB / 32];
  const unsigned wave = threadIdx.x >> 5;
  if ((threadIdx.x & 31u) == 0) wsum[wave] = make_float2(lt, ct);
  __syncthreads();
  if (threadIdx.x == 0) {
    float L = 0.0f, Cn = 0.0f;
#pragma unroll
    for (int w = 0; w < TPB / 32; ++w) {
      L += wsum[w].x;
      Cn += wsum[w].y;
    }
    partials[blockIdx.x] = make_float2(L, Cn);
  }
}

__global__ __launch_bounds__(TPB) void lms_finalize_kernel(
    const float2* __restrict__ partials, float* __restrict__ out) {
  float L = 0.0f, Cn = 0.0f;
  for (unsigned i = threadIdx.x; i < NBLOCKS; i += TPB) {
    const float2 p = partials[i];
    L += p.x;
    Cn += p.y;
  }
  float lt, ct;
  wave_reduce_pair_wmma(L, Cn, &lt, &ct);

  __shared__ float2 wsum[TPB / 32];
  const unsigned wave = threadIdx.x >> 5;
  if ((threadIdx.x & 31u) == 0) wsum[wave] = make_float2(lt, ct);
  __syncthreads();
  if (threadIdx.x == 0) {
    float Lt = 0.0f, Ct = 0.0f;
#pragma unroll
    for (int w = 0; w < TPB / 32; ++w) {
      Lt += wsum[w].x;
      Ct += wsum[w].y;
    }
    out[0] = Lt / Ct;
  }
}

extern "C" void kernel_launch(void* const* d_in, const int* in_sizes, int n_in,
                              void* d_out, int out_size, void* d_ws,
                              size_t ws_size, hipStream_t stream) {
  const float* logits = (const float*)d_in[0];
  const int* labels = (const int*)d_in[1];
  float2* partials = (float2*)d_ws;   // 2048 * 8B = 16 KB scratch

  lms_pixel_kernel<<<NBLOCKS, TPB, 0, stream>>>(logits, labels, partials);
  lms_finalize_kernel<<<1, TPB, 0, stream>>>(partials, (float*)d_out);
}